// SAGEConv_69423851373029
// MI455X (gfx1250) — compile-verified
//
#include <hip/hip_runtime.h>
#include <hip/hip_bf16.h>

typedef __attribute__((ext_vector_type(2))) float v2f;
typedef __attribute__((ext_vector_type(8))) float v8f;

#define D_FEAT 128

// ---------------------------------------------------------------------------
// Kernel 0: zero agg[N*128] and deg[N]
// ---------------------------------------------------------------------------
__global__ void sage_zero_kernel(float* buf, int total) {
    int tid = blockIdx.x * blockDim.x + threadIdx.x;
    if (tid < total) buf[tid] = 0.0f;
}

// ---------------------------------------------------------------------------
// Kernel 1: Xp = feat @ W_src^T   (NT GEMM, fp32 WMMA 16x16x4)
// feat [N,128] row-major, W [128,128] row-major ([out,in] -> B[k][j]=W[j][k])
// Block: (32,8) = 8 waves; wave w -> 16x16 tile at (rows blockIdx.x*16, cols w*16)
// ---------------------------------------------------------------------------
__global__ __launch_bounds__(256) void sage_gemm_nt_kernel(
    const float* __restrict__ A, const float* __restrict__ W,
    float* __restrict__ out, int N) {
    const int lane = threadIdx.x;            // 0..31 (wave32)
    const int wv   = threadIdx.y;            // 0..7
    const int row0 = blockIdx.x * 16;
    const int col0 = wv * 16;
    const int half = lane >> 4;              // 0: K={0,1}, 1: K={2,3}
    const int lr   = lane & 15;
    int arow = row0 + lr; if (arow >= N) arow = N - 1;   // clamp (N mult of 16 anyway)
    const int brow = col0 + lr;
    const int koff = half * 2;

    v8f acc = {};
    const float* ap = A + (size_t)arow * D_FEAT + koff;
    const float* bp = W + (size_t)brow * D_FEAT + koff;
#pragma unroll
    for (int k = 0; k < D_FEAT; k += 4) {
        v2f a = *(const v2f*)(ap + k);
        v2f b = *(const v2f*)(bp + k);
        acc = __builtin_amdgcn_wmma_f32_16x16x4_f32(
            /*neg_a=*/false, a, /*neg_b=*/false, b,
            /*c_mod=*/(short)0, acc, /*reuse_a=*/false, /*reuse_b=*/false);
    }
    // C/D layout: lanes 0-15 rows 0..7, lanes 16-31 rows 8..15; col = lr
    const int ccol  = col0 + lr;
    const int rbase = row0 + half * 8;
#pragma unroll
    for (int r = 0; r < 8; ++r) {
        int row = rbase + r;
        if (row < N) out[(size_t)row * D_FEAT + ccol] = acc[r];
    }
}

// ---------------------------------------------------------------------------
// Kernel 2: edge scatter. One wave per edge: 32 lanes x float4 = 128 floats.
// agg[dst] += Xp[src];  deg[dst] += 1  (lane 0)
// ---------------------------------------------------------------------------
__global__ __launch_bounds__(256) void sage_scatter_kernel(
    const float* __restrict__ Xp, const int* __restrict__ src,
    const int* __restrict__ dst, float* __restrict__ agg,
    float* __restrict__ deg, int E) {
    const int e = blockIdx.x * blockDim.y + threadIdx.y;
    if (e >= E) return;
    const int lane = threadIdx.x;
    const int s = src[e];
    const int d = dst[e];
    const float4 v = *(const float4*)(Xp + (size_t)s * D_FEAT + lane * 4);
    float* p = agg + (size_t)d * D_FEAT + lane * 4;
    atomicAdd(p + 0, v.x);
    atomicAdd(p + 1, v.y);
    atomicAdd(p + 2, v.z);
    atomicAdd(p + 3, v.w);
    if (lane == 0) atomicAdd(deg + d, 1.0f);
}

// ---------------------------------------------------------------------------
// Kernel 3: in-place normalize: agg -> neigh = agg/max(deg,1) + (deg>0)*b_src
// ---------------------------------------------------------------------------
__global__ void sage_norm_kernel(float* __restrict__ agg,
                                 const float* __restrict__ deg,
                                 const float* __restrict__ b_src, int N) {
    int tid = blockIdx.x * blockDim.x + threadIdx.x;
    if (tid >= N * D_FEAT) return;
    int v = tid >> 7;       // /128
    int j = tid & 127;
    float dg = deg[v];
    float m = fmaxf(dg, 1.0f);
    float b = (dg > 0.0f) ? b_src[j] : 0.0f;
    agg[tid] = agg[tid] / m + b;
}

// ---------------------------------------------------------------------------
// Kernel 4: h = feat @ Wl^T + neigh @ Wr^T + b_fc   (W_fc [128,256] row-major)
// Fused K=256 accumulate into one WMMA C tile.
// ---------------------------------------------------------------------------
__global__ __launch_bounds__(256) void sage_gemm2_kernel(
    const float* __restrict__ feat, const float* __restrict__ neigh,
    const float* __restrict__ Wfc, const float* __restrict__ b_fc,
    float* __restrict__ out, int N) {
    const int lane = threadIdx.x;
    const int wv   = threadIdx.y;
    const int row0 = blockIdx.x * 16;
    const int col0 = wv * 16;
    const int half = lane >> 4;
    const int lr   = lane & 15;
    int arow = row0 + lr; if (arow >= N) arow = N - 1;
    const int brow = col0 + lr;
    const int koff = half * 2;

    v8f acc = {};
    const float* ap0 = feat  + (size_t)arow * D_FEAT + koff;
    const float* ap1 = neigh + (size_t)arow * D_FEAT + koff;
    const float* bp  = Wfc   + (size_t)brow * (2 * D_FEAT) + koff;
#pragma unroll
    for (int k = 0; k < D_FEAT; k += 4) {          // first half: feat x Wl
        v2f a = *(const v2f*)(ap0 + k);
        v2f b = *(const v2f*)(bp + k);
        acc = __builtin_amdgcn_wmma_f32_16x16x4_f32(
            false, a, false, b, (short)0, acc, false, false);
    }
#pragma unroll
    for (int k = 0; k < D_FEAT; k += 4) {          // second half: neigh x Wr
        v2f a = *(const v2f*)(ap1 + k);
        v2f b = *(const v2f*)(bp + D_FEAT + k);
        acc = __builtin_amdgcn_wmma_f32_16x16x4_f32(
            false, a, false, b, (short)0, acc, false, false);
    }
    const int ccol  = col0 + lr;
    const int rbase = row0 + half * 8;
    const float bias = b_fc[ccol];
#pragma unroll
    for (int r = 0; r < 8; ++r) {
        int row = rbase + r;
        if (row < N) out[(size_t)row * D_FEAT + ccol] = acc[r] + bias;
    }
}

// ---------------------------------------------------------------------------
// Launch
// inputs: 0=feat[N,128] 1=W_src[128,128] 2=b_src[128] 3=W_fc[128,256]
//         4=b_fc[128] 5=src[E] 6=dst[E]
// ---------------------------------------------------------------------------
extern "C" void kernel_launch(void* const* d_in, const int* in_sizes, int n_in,
                              void* d_out, int out_size, void* d_ws, size_t ws_size,
                              hipStream_t stream) {
    const float* feat  = (const float*)d_in[0];
    const float* W_src = (const float*)d_in[1];
    const float* b_src = (const float*)d_in[2];
    const float* W_fc  = (const float*)d_in[3];
    const float* b_fc  = (const float*)d_in[4];
    const int*   src   = (const int*)d_in[5];
    const int*   dst   = (const int*)d_in[6];

    const int N = in_sizes[0] / D_FEAT;   // 10000
    const int E = in_sizes[5];            // 640000

    float* ws  = (float*)d_ws;
    float* Xp  = ws;                       // N*128
    float* agg = ws + (size_t)N * D_FEAT;  // N*128 (becomes neigh in-place)
    float* deg = agg + (size_t)N * D_FEAT; // N

    float* out = (float*)d_out;

    // 0) zero agg + deg (every call: harness does not re-zero between replays)
    {
        int total = N * D_FEAT + N;
        sage_zero_kernel<<<(total + 255) / 256, 256, 0, stream>>>(agg, total);
    }
    // 1) Xp = feat @ W_src^T
    {
        dim3 blk(32, 8);
        sage_gemm_nt_kernel<<<(N + 15) / 16, blk, 0, stream>>>(feat, W_src, Xp, N);
    }
    // 2) scatter-add along edges
    {
        dim3 blk(32, 8);
        sage_scatter_kernel<<<(E + 7) / 8, blk, 0, stream>>>(Xp, src, dst, agg, deg, E);
    }
    // 3) normalize to mean + fold b_src
    {
        int total = N * D_FEAT;
        sage_norm_kernel<<<(total + 255) / 256, 256, 0, stream>>>(agg, deg, b_src, N);
    }
    // 4) h = [feat | neigh] @ W_fc^T + b_fc
    {
        dim3 blk(32, 8);
        sage_gemm2_kernel<<<(N + 15) / 16, blk, 0, stream>>>(feat, agg, W_fc, b_fc, out, N);
    }
}